// MipRayMarcher2_83288005804168
// MI455X (gfx1250) — compile-verified
//
#include <hip/hip_runtime.h>
#include <stdint.h>

// ---------------------------------------------------------------------------
// MipRayMarcher2 (EG3D) for MI455X / gfx1250.
// Bandwidth-bound streaming scan: ~152MB traffic -> ~6.5us @ 23.3 TB/s.
// CDNA5 path: double-buffered GLOBAL_LOAD_ASYNC_TO_LDS_B128 staging with
// s_wait_asynccnt pipelining (ASYNCcnt), wave32 shuffle + atomic min/max
// reduction for the global depth clamp. WMMA is not applicable (per-ray
// order-dependent scan, no shared contraction).
// NOTE: raymarch_kernel is defined first so the disasm snippet shows it.
// ---------------------------------------------------------------------------

#define NSAMP   96
#define NINT    95
#define CS      8            // samples per staged chunk
#define NCHUNK  12           // 12 * 8 = 96 samples
#define RPB     128          // rays (threads) per block
#define RSTRIDE 40           // dwords/ray/chunk: 24 color + 8 density + 8 depth

typedef int v4i __attribute__((vector_size(16)));

#if defined(__HIP_DEVICE_COMPILE__) && __has_builtin(__builtin_amdgcn_global_load_async_to_lds_b128)
#define HAVE_ASYNC 1
#else
#define HAVE_ASYNC 0
#endif

#if HAVE_ASYNC
// 16B global -> LDS DMA, per-lane, tracked by ASYNCcnt.
// Toolchain signature (from diagnostics): params are v4i* (generic AS).
#define ASYNC_CP16(dst, src)                                                   \
  __builtin_amdgcn_global_load_async_to_lds_b128((v4i*)(src), (v4i*)(dst), 0, 0)
#if __has_builtin(__builtin_amdgcn_s_wait_asynccnt)
#define WAIT_ASYNC(n)                                                          \
  do { __builtin_amdgcn_s_wait_asynccnt(n); asm volatile("" ::: "memory"); } while (0)
#else
#define WAIT_ASYNC(n) asm volatile("s_wait_asynccnt " #n ::: "memory")
#endif
#else
// Fallback: synchronous 16B copy through generic pointers (still B128 loads).
#define ASYNC_CP16(dst, src) (*(float4*)(dst) = *(const float4*)(src))
#define WAIT_ASYNC(n) do { } while (0)
#endif

// Stage one chunk (8 samples) of one ray into LDS: 10 x b128.
__device__ __forceinline__ void stage_chunk(float* dst, const float* gcol,
                                            const float* gden, const float* gdep,
                                            int c) {
  const float* pc = gcol + c * (CS * 3);  // 24 floats = 96B  (6 x b128)
  const float* pn = gden + c * CS;        // 8 floats  = 32B  (2 x b128)
  const float* pz = gdep + c * CS;        // 8 floats  = 32B  (2 x b128)
#pragma unroll
  for (int p = 0; p < 6; ++p) ASYNC_CP16(dst + 4 * p, pc + 4 * p);
  ASYNC_CP16(dst + 24, pn);
  ASYNC_CP16(dst + 28, pn + 4);
  ASYNC_CP16(dst + 32, pz);
  ASYNC_CP16(dst + 36, pz + 4);
}

// --- main pass: the ray-march scan (defined first -> shown in disasm) ------
__global__ __launch_bounds__(RPB) void raymarch_kernel(
    const float* __restrict__ colors, const float* __restrict__ densities,
    const float* __restrict__ depths, const unsigned* __restrict__ mm,
    float* __restrict__ out_rgb, float* __restrict__ out_depth,
    float* __restrict__ out_w, int nRays) {
  __shared__ float lds[2][RPB * RSTRIDE];  // 2 x 20KB double buffer
  const int t = threadIdx.x;
  const long ray = (long)blockIdx.x * RPB + t;
  if (ray >= nRays) return;

  const float* gcol = colors + ray * (long)(NSAMP * 3);
  const float* gden = densities + ray * (long)NSAMP;
  const float* gdep = depths + ray * (long)NSAMP;
  float* b0 = &lds[0][t * RSTRIDE];
  float* b1 = &lds[1][t * RSTRIDE];
  float* wout = out_w + ray * (long)NINT;

  stage_chunk(b0, gcol, gden, gdep, 0);

  float T = 1.0f;
  float sr = 0.f, sg = 0.f, sb = 0.f, sw = 0.f, swd = 0.f;
  float pc0 = 0.f, pc1 = 0.f, pc2 = 0.f, pd = 0.f, pz = 0.f;

  for (int c = 0; c < NCHUNK; ++c) {
    float* cur = (c & 1) ? b1 : b0;
    if (c + 1 < NCHUNK) {
      stage_chunk((c & 1) ? b0 : b1, gcol, gden, gdep, c + 1);
      WAIT_ASYNC(10);  // 10 newer in flight; older chunk (in-order) is done
    } else {
      WAIT_ASYNC(0);
    }

    int s0 = 0;
    if (c == 0) {  // seed "previous sample" with sample 0
      pc0 = cur[0]; pc1 = cur[1]; pc2 = cur[2];
      pd = cur[24]; pz = cur[32];
      s0 = 1;
    }
#pragma unroll
    for (int s = 0; s < CS; ++s) {
      if (s < s0) continue;
      float cc0 = cur[3 * s], cc1 = cur[3 * s + 1], cc2 = cur[3 * s + 2];
      float cd = cur[24 + s], cz = cur[32 + s];

      float delta = cz - pz;
      float x = 0.5f * (pd + cd) - 1.0f;
      // softplus(x) = max(x,0) + log1p(exp(-|x|))  (matches jax.nn.softplus)
      float sp = fmaxf(x, 0.0f) + log1pf(expf(-fabsf(x)));
      float alpha = 1.0f - expf(-sp * delta);
      float w = alpha * T;

      wout[c * CS + s - 1] = w;
      sr = fmaf(w, 0.5f * (pc0 + cc0), sr);
      sg = fmaf(w, 0.5f * (pc1 + cc1), sg);
      sb = fmaf(w, 0.5f * (pc2 + cc2), sb);
      sw += w;
      swd = fmaf(w, 0.5f * (pz + cz), swd);
      T = T * (1.0f - alpha + 1e-10f);

      pc0 = cc0; pc1 = cc1; pc2 = cc2; pd = cd; pz = cz;
    }
  }

  out_rgb[ray * 3 + 0] = sr * 2.0f - 1.0f;
  out_rgb[ray * 3 + 1] = sg * 2.0f - 1.0f;
  out_rgb[ray * 3 + 2] = sb * 2.0f - 1.0f;

  float gmin = __uint_as_float(mm[0]);
  float gmax = __uint_as_float(mm[1]);
  float dc = swd / sw;                 // 0/0 -> NaN -> inf -> clamps to gmax
  if (__builtin_isnan(dc)) dc = gmax;
  dc = fminf(fmaxf(dc, gmin), gmax);
  out_depth[ray] = dc;
}

// --- pass 1: init global min/max slots in workspace ------------------------
__global__ void init_minmax_kernel(unsigned* mm) {
  mm[0] = 0x7F800000u;  // +inf bits (depths are positive -> uint order == float order)
  mm[1] = 0x00000000u;
}

// --- pass 2: global min/max of depths (sorted along S: first/last sample) --
__global__ void minmax_kernel(const float* __restrict__ depths, unsigned* mm,
                              int nRays) {
  int i = blockIdx.x * blockDim.x + threadIdx.x;
  float lo = __builtin_inff(), hi = -__builtin_inff();
  if (i < nRays) {
    lo = depths[(size_t)i * NSAMP];
    hi = depths[(size_t)i * NSAMP + (NSAMP - 1)];
  }
#pragma unroll
  for (int off = 16; off > 0; off >>= 1) {       // wave32 reduction
    lo = fminf(lo, __shfl_down(lo, off));
    hi = fmaxf(hi, __shfl_down(hi, off));
  }
  if ((threadIdx.x & 31) == 0) {
    atomicMin(&mm[0], __float_as_uint(lo));
    atomicMax(&mm[1], __float_as_uint(hi));
  }
}

extern "C" void kernel_launch(void* const* d_in, const int* in_sizes, int n_in,
                              void* d_out, int out_size, void* d_ws, size_t ws_size,
                              hipStream_t stream) {
  const float* colors    = (const float*)d_in[0];   // [B,R,S,3]
  const float* densities = (const float*)d_in[1];   // [B,R,S,1]
  const float* depths    = (const float*)d_in[2];   // [B,R,S,1]
  const int nRays = in_sizes[1] / NSAMP;            // B*R = 65536

  float* out       = (float*)d_out;
  float* out_rgb   = out;                           // [B,R,3]
  float* out_depth = out + (size_t)nRays * 3;       // [B,R,1]
  float* out_w     = out_depth + nRays;             // [B,R,95,1]
  unsigned* mm     = (unsigned*)d_ws;

  hipLaunchKernelGGL(init_minmax_kernel, dim3(1), dim3(1), 0, stream, mm);
  hipLaunchKernelGGL(minmax_kernel, dim3((nRays + 255) / 256), dim3(256), 0,
                     stream, depths, mm, nRays);
  hipLaunchKernelGGL(raymarch_kernel, dim3((nRays + RPB - 1) / RPB), dim3(RPB),
                     0, stream, colors, densities, depths, mm, out_rgb,
                     out_depth, out_w, nRays);
}